// DynamicVoxelization_38903813767723
// MI455X (gfx1250) — compile-verified
//
#include <hip/hip_runtime.h>
#include <hip/hip_bf16.h>

// ---------------- problem constants (from reference) ----------------
#define VOXSZ 0.1f
#define XMIN_ (-20.0f)
#define YMIN_ (-20.0f)
#define ZMIN_ (-2.0f)
#define XMAX_ (20.0f)
#define YMAX_ (20.0f)
#define ZMAX_ (3.0f)
#define GX_ 400
#define GY_ 400
#define P_  32
#define K_  60000
#define B_  4
#define NCELL (B_ * GX_ * GY_)      // 640000 dense cells

// ---------------- tiling for async global->LDS staging ----------------
#define TILE_PTS    256                     // points per tile (== block size)
#define PT_BYTES    20                      // 5 f32 per point
#define TILE_BYTES  (TILE_PTS * PT_BYTES)   // 5120 B
#define TILE_CHUNKS (TILE_BYTES / 16)       // 320 x B128 chunks

#define SCAN_TPB        256
#define CELLS_PER_BLOCK (SCAN_TPB * 4)      // 1024 cells per scan block

// ---------------- CDNA5 async-copy helpers ----------------
typedef int v4i_t __attribute__((ext_vector_type(4)));
typedef __attribute__((address_space(1))) v4i_t* gv4_ptr;
typedef __attribute__((address_space(3))) v4i_t* lv4_ptr;

__device__ __forceinline__ void async_copy16(const void* g, void* l) {
#if __has_builtin(__builtin_amdgcn_global_load_async_to_lds_b128)
  // builtin signature (from hipcc diagnostic): (int4 AS1*, int4 AS3*, Imm, Imm)
  __builtin_amdgcn_global_load_async_to_lds_b128(
      (gv4_ptr)g, (lv4_ptr)l, /*offset=*/0, /*cpol=*/0);
#else
  // VGLOBAL async encoding: VDST = LDS byte offset VGPR, VADDR = 64-bit address
  unsigned lds_off = (unsigned)(unsigned long long)l;
  asm volatile("global_load_async_to_lds_b128 %0, %1, off"
               :: "v"(lds_off), "v"(g) : "memory");
#endif
}

__device__ __forceinline__ void wait_async0() {
#if __has_builtin(__builtin_amdgcn_s_wait_asynccnt)
  __builtin_amdgcn_s_wait_asynccnt(0);
#else
  asm volatile("s_wait_asynccnt 0" ::: "memory");
#endif
}

__device__ __forceinline__ int iminD(int a, int b) { return a < b ? a : b; }
__device__ __forceinline__ int imaxD(int a, int b) { return a > b ? a : b; }

// stage one tile of raw point bytes into LDS via async B128 DMA
__device__ __forceinline__ void stage_tile(const float* __restrict__ pts,
                                           float* lds, int tile, long totalBytes) {
  const int t = threadIdx.x;
  long tileByte = (long)tile * TILE_BYTES;
#pragma unroll
  for (int pass = 0; pass < 2; ++pass) {
    int c = pass * 256 + t;                 // chunk index within tile
    long gb = tileByte + (long)c * 16;
    if (c < TILE_CHUNKS && gb < totalBytes) {
      async_copy16((const char*)pts + gb, (char*)lds + (size_t)c * 16);
    }
  }
  wait_async0();      // this wave's async ops landed
  __syncthreads();    // all waves' chunks visible
}

__device__ __forceinline__ int point_flat_id(float b, float x, float y, float z,
                                             bool* maskOut) {
  bool mask = (x >= XMIN_) & (x < XMAX_) & (y >= YMIN_) & (y < YMAX_) &
              (z >= ZMIN_) & (z < ZMAX_);
  *maskOut = mask;
  int bi = (int)b;
  int xi = iminD(imaxD((int)((x - XMIN_) / VOXSZ), 0), GX_ - 1);
  int yi = iminD(imaxD((int)((y - YMIN_) / VOXSZ), 0), GY_ - 1);
  return bi * (GX_ * GY_) + xi * GY_ + yi;
}

// ---------------- K0: zero outputs + dense counters ----------------
__global__ void vox_zero(float* __restrict__ out, long nOut,
                         int* __restrict__ cnts, long nCnt) {
  long i = (long)blockIdx.x * blockDim.x + threadIdx.x;
  long stride = (long)gridDim.x * blockDim.x;
  long n4 = nOut >> 2;
  float4* out4 = (float4*)out;
  float4 z4 = make_float4(0.f, 0.f, 0.f, 0.f);
  for (long j = i; j < n4; j += stride) out4[j] = z4;
  for (long j = n4 * 4 + i; j < nOut; j += stride) out[j] = 0.f;
  for (long j = i; j < nCnt; j += stride) cnts[j] = 0;
}

// ---------------- K1: per-point counting via dense atomics ----------------
__global__ void vox_count(const float* __restrict__ pts,
                          int* __restrict__ cnt, int nPoints) {
  __shared__ float lds[TILE_BYTES / 4];     // 1280 f32 = one tile
  const int t = threadIdx.x;
  const int tile = blockIdx.x;
  long totalBytes = (long)nPoints * PT_BYTES;
  stage_tile(pts, lds, tile, totalBytes);

  int pi = tile * TILE_PTS + t;
  if (pi < nPoints) {
    float b = lds[t * 5 + 0], x = lds[t * 5 + 1];
    float y = lds[t * 5 + 2], z = lds[t * 5 + 3];
    bool mask;
    int flat = point_flat_id(b, x, y, z, &mask);
    if (mask) atomicAdd(&cnt[flat], 1);
  }
}

// ---------------- K2: per-block occupancy scan ----------------
__global__ void vox_scan_blocks(const int* __restrict__ cnt,
                                int* __restrict__ rowmap,
                                int* __restrict__ blocksums, int ncell) {
  __shared__ int sdata[SCAN_TPB];
  const int t = threadIdx.x;
  const int base = blockIdx.x * CELLS_PER_BLOCK + t * 4;
  int f[4]; int s = 0;
#pragma unroll
  for (int k = 0; k < 4; ++k) {
    int i = base + k;
    f[k] = (i < ncell && cnt[i] > 0) ? 1 : 0;
    s += f[k];
  }
  sdata[t] = s;
  __syncthreads();
  for (int off = 1; off < SCAN_TPB; off <<= 1) {   // Hillis-Steele inclusive
    int v = (t >= off) ? sdata[t - off] : 0;
    __syncthreads();
    sdata[t] += v;
    __syncthreads();
  }
  int incl = sdata[t];
  int p = incl - s;                                 // exclusive for this thread
#pragma unroll
  for (int k = 0; k < 4; ++k) {
    int i = base + k;
    if (i < ncell) rowmap[i] = p;                   // local exclusive prefix
    p += f[k];
  }
  if (t == SCAN_TPB - 1) blocksums[blockIdx.x] = incl;
}

// ---------------- K3: scan the 625 block sums (single block) ----------------
__global__ void vox_scan_sums(int* __restrict__ blocksums,
                              int* __restrict__ numUniq, int nblocks) {
  __shared__ int sdata[1024];
  const int t = threadIdx.x;
  int v = (t < nblocks) ? blocksums[t] : 0;
  sdata[t] = v;
  __syncthreads();
  for (int off = 1; off < 1024; off <<= 1) {
    int u = (t >= off) ? sdata[t - off] : 0;
    __syncthreads();
    sdata[t] += u;
    __syncthreads();
  }
  if (t < nblocks) blocksums[t] = sdata[t] - v;     // exclusive
  if (t == 1023) *numUniq = sdata[1023];
}

// ---------------- K4: finalize ranks, counts, coords, fill rows ----------------
__global__ void vox_finalize(const int* __restrict__ cnt,
                             int* __restrict__ rowmap,
                             const int* __restrict__ blocksums,
                             const int* __restrict__ numUniq,
                             float* __restrict__ counts_out,
                             float* __restrict__ coors_out, int ncell) {
  int i = blockIdx.x * blockDim.x + threadIdx.x;
  if (i < ncell) {
    int c = cnt[i];
    int rank = -1;
    if (c > 0) {
      rank = rowmap[i] + blocksums[i / CELLS_PER_BLOCK];
      if (rank < K_) {
        counts_out[rank] = (float)iminD(c, P_);
        int b = i / (GX_ * GY_);
        int rem = i % (GX_ * GY_);
        coors_out[rank * 3 + 0] = (float)b;
        coors_out[rank * 3 + 1] = (float)(rem / GY_);
        coors_out[rank * 3 + 2] = (float)(rem % GY_);
      } else {
        rank = -1;                                  // overflow beyond K dropped
      }
    }
    rowmap[i] = rank;
  }
  if (i < K_) {
    int nu = *numUniq;
    if (i >= nu) {                                  // sentinel fill rows
      counts_out[i] = 0.0f;
      coors_out[i * 3 + 0] = (float)B_;             // 640000/(GX*GY) = 4
      coors_out[i * 3 + 1] = 0.0f;
      coors_out[i * 3 + 2] = 0.0f;
    }
  }
}

// ---------------- K5: scatter point features into voxel slots ----------------
__global__ void vox_scatter(const float* __restrict__ pts,
                            const int* __restrict__ rowmap,
                            int* __restrict__ cnt2,
                            float* __restrict__ voxels, int nPoints) {
  __shared__ float lds[TILE_BYTES / 4];
  const int t = threadIdx.x;
  const int tile = blockIdx.x;
  long totalBytes = (long)nPoints * PT_BYTES;
  stage_tile(pts, lds, tile, totalBytes);

  int pi = tile * TILE_PTS + t;
  if (pi < nPoints) {
    float b = lds[t * 5 + 0], x = lds[t * 5 + 1];
    float y = lds[t * 5 + 2], z = lds[t * 5 + 3], r = lds[t * 5 + 4];
    bool mask;
    int flat = point_flat_id(b, x, y, z, &mask);
    if (mask) {
      int row = rowmap[flat];
      int slot = atomicAdd(&cnt2[flat], 1);
      if (row >= 0 && slot < P_) {
        float4 v = make_float4(x, y, z, r);
        *(float4*)&voxels[((long)row * P_ + slot) * 4] = v;   // B128 store
      }
    }
  }
}

// ---------------- launch ----------------
extern "C" void kernel_launch(void* const* d_in, const int* in_sizes, int n_in,
                              void* d_out, int out_size, void* d_ws, size_t ws_size,
                              hipStream_t stream) {
  const float* pts = (const float*)d_in[0];
  const int nPoints = in_sizes[0] / 5;

  float* out = (float*)d_out;
  float* voxels = out;                               // [K, P, 4]
  float* counts = out + (long)K_ * P_ * 4;           // [K]
  float* coors  = counts + K_;                       // [K, 3]
  const long nOut = (long)K_ * P_ * 4 + K_ + (long)K_ * 3;

  int* cnt       = (int*)d_ws;                       // [NCELL]
  int* cnt2      = cnt + NCELL;                      // [NCELL]
  int* rowmap    = cnt2 + NCELL;                     // [NCELL]
  int* blocksums = rowmap + NCELL;                   // [<=1024]
  int* numUniq   = blocksums + 1024;                 // [1]

  const long totalBytes = (long)nPoints * PT_BYTES;
  const int nChunks = (int)((totalBytes + 15) / 16);
  const int nTiles  = (nChunks + TILE_CHUNKS - 1) / TILE_CHUNKS;
  const int nScanBlocks = (NCELL + CELLS_PER_BLOCK - 1) / CELLS_PER_BLOCK; // 625

  vox_zero<<<2048, 256, 0, stream>>>(out, nOut, cnt, 2L * NCELL);
  vox_count<<<nTiles, TILE_PTS, 0, stream>>>(pts, cnt, nPoints);
  vox_scan_blocks<<<nScanBlocks, SCAN_TPB, 0, stream>>>(cnt, rowmap, blocksums, NCELL);
  vox_scan_sums<<<1, 1024, 0, stream>>>(blocksums, numUniq, nScanBlocks);
  vox_finalize<<<(NCELL + 255) / 256, 256, 0, stream>>>(cnt, rowmap, blocksums,
                                                        numUniq, counts, coors, NCELL);
  vox_scatter<<<nTiles, TILE_PTS, 0, stream>>>(pts, rowmap, cnt2, voxels, nPoints);
}